// PhysicsEmbedding_9895604650678
// MI455X (gfx1250) — compile-verified
//
#include <hip/hip_runtime.h>
#include <math.h>

typedef __attribute__((ext_vector_type(2))) float v2f;
typedef __attribute__((ext_vector_type(4))) float v4f;
typedef __attribute__((ext_vector_type(8))) float v8f;

#define B_    32
#define S_    4096
#define D_    1024
#define H_    256
#define CHUNK 128

// ---------------------------------------------------------------------------
// Kernel A: 16-bin DFT magnitudes per (batch, channel).
// F[b, ch*16 + k] = | sum_n pattern[b,n,ch] * exp(-2*pi*i*k*n/S) |
// Exact phase reduction: (k*n) mod 4096 is exact in int, angle in [0, 2pi).
// ---------------------------------------------------------------------------
__global__ void __launch_bounds__(256) dft16_kernel(const float* __restrict__ pattern,
                                                    float* __restrict__ F) {
    const int blk = blockIdx.x;          // b*32 + ch*16 + k
    const int b  = blk >> 5;
    const int ch = (blk >> 4) & 1;
    const int k  = blk & 15;
    const int t  = threadIdx.x;

    float re = 0.f, im = 0.f;
    for (int n = t; n < S_; n += 256) {
        float x = pattern[(b * S_ + n) * 2 + ch];
        int   m = (k * n) & (S_ - 1);
        float ang = -6.283185307179586f * ((float)m / (float)S_);
        float s, c;
        __sincosf(ang, &s, &c);
        re += x * c;
        im += x * s;
    }
    __shared__ float sre[256], sim[256];
    sre[t] = re; sim[t] = im;
    __syncthreads();
    for (int off = 128; off > 0; off >>= 1) {
        if (t < off) { sre[t] += sre[t + off]; sim[t] += sim[t + off]; }
        __syncthreads();
    }
    if (t == 0)
        F[b * 32 + ch * 16 + k] = sqrtf(sre[0] * sre[0] + sim[0] * sim[0]);
}

// ---------------------------------------------------------------------------
// Kernel B: fold the 2-wide branches and all biases.
//   Ep[d,:] = Wo[d, 0:256]   @ Wp     (1024x2)
//   Ev[d,:] = Wo[d, 256:512] @ Wv
//   Ea[d,:] = Wo[d, 512:768] @ Wa
//   beff[d] = bo[d] + Wo[d,:] . [bp;bv;ba;bf]
// ---------------------------------------------------------------------------
__global__ void __launch_bounds__(256) fold_small_kernel(
    const float* __restrict__ Wp, const float* __restrict__ bp,
    const float* __restrict__ Wv, const float* __restrict__ bv,
    const float* __restrict__ Wa, const float* __restrict__ ba,
    const float* __restrict__ bf,
    const float* __restrict__ Wo, const float* __restrict__ bo,
    float* __restrict__ Ep, float* __restrict__ Ev, float* __restrict__ Ea,
    float* __restrict__ beff) {
    const int d = blockIdx.x * blockDim.x + threadIdx.x;
    if (d >= D_) return;
    const float* wo = Wo + (size_t)d * D_;
    float ep0 = 0, ep1 = 0, ev0 = 0, ev1 = 0, ea0 = 0, ea1 = 0, bs = 0;
    for (int k = 0; k < H_; ++k) {
        float w0 = wo[k], w1 = wo[H_ + k], w2 = wo[2 * H_ + k], w3 = wo[3 * H_ + k];
        ep0 += w0 * Wp[k * 2];  ep1 += w0 * Wp[k * 2 + 1];
        ev0 += w1 * Wv[k * 2];  ev1 += w1 * Wv[k * 2 + 1];
        ea0 += w2 * Wa[k * 2];  ea1 += w2 * Wa[k * 2 + 1];
        bs  += w0 * bp[k] + w1 * bv[k] + w2 * ba[k] + w3 * bf[k];
    }
    Ep[d * 2] = ep0; Ep[d * 2 + 1] = ep1;
    Ev[d * 2] = ev0; Ev[d * 2 + 1] = ev1;
    Ea[d * 2] = ea0; Ea[d * 2 + 1] = ea1;
    beff[d] = bo[d] + bs;
}

// ---------------------------------------------------------------------------
// Kernel C (WMMA, fp32): Ef = Wo[:, 768:1024] @ Wf   (M=1024, N=32, K=256)
// One wave per 16x16 tile, K stepped by 4 with V_WMMA_F32_16X16X4_F32.
// A 16x4 layout: lanes 0-15 hold K={0,1} (VGPR0,1), lanes 16-31 hold K={2,3}.
// B 4x16 layout mirrors A (low lanes = first half of K range).
// C/D 16x16 f32: VGPR r -> M=r (lanes 0-15), M=8+r (lanes 16-31).
// ---------------------------------------------------------------------------
__global__ void __launch_bounds__(32) ef_wmma_kernel(const float* __restrict__ Wo,
                                                     const float* __restrict__ Wf,
                                                     float* __restrict__ Ef) {
    const int tile = blockIdx.x;         // 0..127 : 64 M-tiles x 2 N-tiles
    const int m0   = (tile >> 1) << 4;
    const int n0   = (tile & 1) << 4;
    const int lane = threadIdx.x & 31;
    const int half = lane >> 4;          // 0: K pair {0,1}, 1: K pair {2,3}
    const int lm   = lane & 15;

    v8f acc = {};
    for (int k = 0; k < H_; k += 4) {
        const int ka = k + half * 2;
        v2f a, b;
        a.x = Wo[(size_t)(m0 + lm) * D_ + 3 * H_ + ka];
        a.y = Wo[(size_t)(m0 + lm) * D_ + 3 * H_ + ka + 1];
        b.x = Wf[(ka)     * 32 + n0 + lm];
        b.y = Wf[(ka + 1) * 32 + n0 + lm];
        acc = __builtin_amdgcn_wmma_f32_16x16x4_f32(
            /*neg_a=*/false, a, /*neg_b=*/false, b,
            /*c_mod=*/(short)0, acc, /*reuse_a=*/false, /*reuse_b=*/false);
    }
#pragma unroll
    for (int r = 0; r < 8; ++r)
        Ef[(m0 + half * 8 + r) * 32 + n0 + lm] = acc[r];
}

// ---------------------------------------------------------------------------
// Kernel D: G[b,d] = F[b,:] . Ef[d,:] + beff[d]     (32 x 1024)
// ---------------------------------------------------------------------------
__global__ void __launch_bounds__(256) g_kernel(const float* __restrict__ F,
                                                const float* __restrict__ Ef,
                                                const float* __restrict__ beff,
                                                float* __restrict__ G) {
    const int idx = blockIdx.x * blockDim.x + threadIdx.x;
    if (idx >= B_ * D_) return;
    const int b = idx >> 10;
    const int d = idx & (D_ - 1);
    const float* f = F + b * 32;
    const float* e = Ef + d * 32;
    float acc = beff[d];
#pragma unroll
    for (int j = 0; j < 32; ++j) acc += f[j] * e[j];
    G[idx] = acc;
}

// ---------------------------------------------------------------------------
// Kernel E: the store-bandwidth-bound stream.
// out[b,s,d] = p.Ep[d] + v.Ev[d] + a.Ea[d] + G[b,d], 6 FMAs / element.
// One block = one (b, 128-timestep chunk); thread t owns d = 4t..4t+3.
// Coefficients live in registers; pattern chunk staged through LDS;
// nontemporal b128 stores (512MB stream, bypass L2 pollution).
// ---------------------------------------------------------------------------
__global__ void __launch_bounds__(256) stream_kernel(
    const float* __restrict__ pattern,
    const float* __restrict__ Ep, const float* __restrict__ Ev,
    const float* __restrict__ Ea, const float* __restrict__ G,
    float* __restrict__ out) {
    const int nchunks = S_ / CHUNK;
    const int b  = blockIdx.x / nchunks;
    const int s0 = (blockIdx.x % nchunks) * CHUNK;
    const int t  = threadIdx.x;
    const int d0 = t * 4;

    // Gather per-thread coefficient registers (rows d0..d0+3 of [1024x2] mats).
    v4f ra, rb;
    ra = *(const v4f*)(Ep + d0 * 2);     rb = *(const v4f*)(Ep + d0 * 2 + 4);
    v4f ep0 = { ra.x, ra.z, rb.x, rb.z }, ep1 = { ra.y, ra.w, rb.y, rb.w };
    ra = *(const v4f*)(Ev + d0 * 2);     rb = *(const v4f*)(Ev + d0 * 2 + 4);
    v4f ev0 = { ra.x, ra.z, rb.x, rb.z }, ev1 = { ra.y, ra.w, rb.y, rb.w };
    ra = *(const v4f*)(Ea + d0 * 2);     rb = *(const v4f*)(Ea + d0 * 2 + 4);
    v4f ea0 = { ra.x, ra.z, rb.x, rb.z }, ea1 = { ra.y, ra.w, rb.y, rb.w };
    v4f g = *(const v4f*)(G + b * D_ + d0);

    // Stage pattern[s0-2 .. s0+CHUNK-1] (both channels) into LDS.
    __shared__ float sp[(CHUNK + 2) * 2];
    for (int i = t; i < (CHUNK + 2) * 2; i += 256) {
        const int gs = s0 - 2 + (i >> 1);
        const int ch = i & 1;
        sp[i] = (gs >= 0) ? pattern[((size_t)b * S_ + gs) * 2 + ch] : 0.f;
    }
    __syncthreads();

    for (int i = 0; i < CHUNK; ++i) {
        const int s = s0 + i;
        const float p0 = sp[(i + 2) * 2], p1 = sp[(i + 2) * 2 + 1];
        const float q0 = sp[(i + 1) * 2], q1 = sp[(i + 1) * 2 + 1];
        const float r0 = sp[i * 2],       r1 = sp[i * 2 + 1];
        // vel[s] = p[s]-p[s-1] (0 at s=0); acc[s] = vel[s]-vel[s-1] (0 at s=0)
        const float v0 = (s == 0) ? 0.f : p0 - q0;
        const float v1 = (s == 0) ? 0.f : p1 - q1;
        const float w0 = (s <= 1) ? 0.f : q0 - r0;   // vel[s-1]
        const float w1 = (s <= 1) ? 0.f : q1 - r1;
        const float a0 = (s == 0) ? 0.f : v0 - w0;
        const float a1 = (s == 0) ? 0.f : v1 - w1;

        v4f o = g;
        o += p0 * ep0; o += p1 * ep1;
        o += v0 * ev0; o += v1 * ev1;
        o += a0 * ea0; o += a1 * ea1;

        __builtin_nontemporal_store(
            o, (v4f*)(out + ((size_t)b * S_ + s) * D_ + d0));
    }
}

// ---------------------------------------------------------------------------
extern "C" void kernel_launch(void* const* d_in, const int* in_sizes, int n_in,
                              void* d_out, int out_size, void* d_ws, size_t ws_size,
                              hipStream_t stream) {
    const float* pattern = (const float*)d_in[0];
    const float* Wp = (const float*)d_in[1];  const float* bp = (const float*)d_in[2];
    const float* Wv = (const float*)d_in[3];  const float* bv = (const float*)d_in[4];
    const float* Wa = (const float*)d_in[5];  const float* ba = (const float*)d_in[6];
    const float* Wf = (const float*)d_in[7];  const float* bf = (const float*)d_in[8];
    const float* Wo = (const float*)d_in[9];  const float* bo = (const float*)d_in[10];
    float* out = (float*)d_out;

    // Workspace layout (floats): F[1024] Ep[2048] Ev[2048] Ea[2048] beff[1024]
    //                            Ef[32768] G[32768]  => 288 KB total
    float* ws   = (float*)d_ws;
    float* F    = ws;
    float* Ep   = ws + 1024;
    float* Ev   = ws + 3072;
    float* Ea   = ws + 5120;
    float* beff = ws + 7168;
    float* Ef   = ws + 8192;
    float* G    = ws + 40960;

    dft16_kernel<<<B_ * 2 * 16, 256, 0, stream>>>(pattern, F);
    fold_small_kernel<<<D_ / 256, 256, 0, stream>>>(Wp, bp, Wv, bv, Wa, ba, bf,
                                                    Wo, bo, Ep, Ev, Ea, beff);
    ef_wmma_kernel<<<(D_ / 16) * 2, 32, 0, stream>>>(Wo, Wf, Ef);
    g_kernel<<<(B_ * D_) / 256, 256, 0, stream>>>(F, Ef, beff, G);
    stream_kernel<<<B_ * (S_ / CHUNK), 256, 0, stream>>>(pattern, Ep, Ev, Ea, G, out);
}